// FDC2_61108794688088
// MI455X (gfx1250) — compile-verified
//
#include <hip/hip_runtime.h>

typedef float v2f __attribute__((ext_vector_type(2)));
typedef float v8f __attribute__((ext_vector_type(8)));

#define LDST 36   // LDS row stride in floats: 144B rows -> 8B/16B friendly, bank-conflict free

__device__ __forceinline__ void st_lds4(float* p, float4 v) {
  // row stride guarantees 8-byte alignment; two b64 stores
  ((v2f*)p)[0] = (v2f){v.x, v.y};
  ((v2f*)p)[1] = (v2f){v.z, v.w};
}

// ---------------------------------------------------------------------------
// K0: zero the zsum accumulator (ws is poisoned / stale between replays)
// ---------------------------------------------------------------------------
__global__ void k_zero(float* __restrict__ zsum) {
  zsum[blockIdx.x * 128 + threadIdx.x] = 0.0f;
}

// ---------------------------------------------------------------------------
// K1: z2p = relu(z2f @ W_proj^T + b_proj), reduced over batch into zsum[1024].
// z2f: [2048, 9408], W_proj: [1024, 9408] (row-major, so B[k][n] = W[n][k]).
// Block: 512 threads = 16 waves; macro-tile 64(M) x 128(N); wave = 16x32 slab.
// ---------------------------------------------------------------------------
__global__ __launch_bounds__(512) void k_proj_colsum(
    const float* __restrict__ A, const float* __restrict__ W,
    const float* __restrict__ bias, float* __restrict__ zsum) {
  __shared__ float As[64 * LDST];
  __shared__ float Bs[128 * LDST];

  const int tid  = threadIdx.x;
  const int lane = tid & 31;
  const int w    = tid >> 5;
  const int l    = lane & 15;
  const int half = lane >> 4;
  const int tm   = (w >> 2) * 16;   // 0..48
  const int tn   = (w & 3) * 32;    // 0..96
  const int m0   = blockIdx.x * 64;
  const int n0   = blockIdx.y * 128;

  // global->LDS assignment
  const int ar = tid >> 3;          // 0..63   (A row)
  const int ac = (tid & 7) << 2;    // 0..28   (A col, float4)
  const int br = tid >> 2;          // 0..127  (B row = n index)
  const int bc = (tid & 3) << 3;    // 0,8,16,24 (8 floats per thread)

  const float* aptr = A + (size_t)(m0 + ar) * 9408 + ac;
  const float* bptr = W + (size_t)(n0 + br) * 9408 + bc;

  v8f acc0 = {}; v8f acc1 = {};

  for (int k0 = 0; k0 < 9408; k0 += 32) {
    const float4 av  = *(const float4*)(aptr + k0);
    const float4 bv0 = *(const float4*)(bptr + k0);
    const float4 bv1 = *(const float4*)(bptr + k0 + 4);
    __syncthreads();
    st_lds4(&As[ar * LDST + ac], av);
    st_lds4(&Bs[br * LDST + bc], bv0);
    st_lds4(&Bs[br * LDST + bc + 4], bv1);
    __syncthreads();

    const float* ap  = &As[(tm + l) * LDST + 2 * half];
    const float* bp0 = &Bs[(tn + l) * LDST + 2 * half];
    const float* bp1 = &Bs[(tn + 16 + l) * LDST + 2 * half];
#pragma unroll
    for (int kk = 0; kk < 32; kk += 4) {
      v2f a  = *(const v2f*)(ap + kk);
      v2f b0 = *(const v2f*)(bp0 + kk);
      v2f b1 = *(const v2f*)(bp1 + kk);
      acc0 = __builtin_amdgcn_wmma_f32_16x16x4_f32(false, a, false, b0,
                                                   (short)0, acc0, false, false);
      acc1 = __builtin_amdgcn_wmma_f32_16x16x4_f32(false, a, false, b1,
                                                   (short)0, acc1, false, false);
    }
  }

  // epilogue: +bias, relu, sum the 8 rows each lane holds, merge M-halves,
  // atomically accumulate per-column sums.
  const int c0 = n0 + tn + l;
  const int c1 = c0 + 16;
  const float b0v = bias[c0];
  const float b1v = bias[c1];
  float s0 = 0.0f, s1 = 0.0f;
#pragma unroll
  for (int r = 0; r < 8; ++r) {
    s0 += fmaxf(acc0[r] + b0v, 0.0f);
    s1 += fmaxf(acc1[r] + b1v, 0.0f);
  }
  s0 += __shfl_xor(s0, 16, 32);   // rows 0-7 + rows 8-15
  s1 += __shfl_xor(s1, 16, 32);
  if (half == 0) {
    atomicAdd(&zsum[c0], s0);
    atomicAdd(&zsum[c1], s1);
  }
}

// ---------------------------------------------------------------------------
// K2: S2adj[c] = zsum . W_fc[c, 2048:3072] + 2048*b_fc[c]    (65 classes)
// ---------------------------------------------------------------------------
__global__ __launch_bounds__(128) void k_s2(
    const float* __restrict__ zsum, const float* __restrict__ Wfc,
    const float* __restrict__ bfc, float* __restrict__ S2adj) {
  __shared__ float red[128];
  const int c = blockIdx.x;                        // 0..64
  const float* wrow = Wfc + (size_t)c * 3072 + 2048;
  float s = 0.0f;
  for (int k = threadIdx.x; k < 1024; k += 128) s += zsum[k] * wrow[k];
  red[threadIdx.x] = s;
  __syncthreads();
  for (int off = 64; off > 0; off >>= 1) {
    if (threadIdx.x < off) red[threadIdx.x] += red[threadIdx.x + off];
    __syncthreads();
  }
  if (threadIdx.x == 0) S2adj[c] = red[0] + 2048.0f * bfc[c];
}

// ---------------------------------------------------------------------------
// K3: out[i,c] = 2048 * (z1 @ W_fc[:, :2048]^T)[i,c] + S2adj[c]
// z1: [2048,2048], W_fc: [65,3072]. Grid (32 m-blocks, 5 n-tiles), 4 waves.
// ---------------------------------------------------------------------------
__global__ __launch_bounds__(128) void k_fc_out(
    const float* __restrict__ z1, const float* __restrict__ Wfc,
    const float* __restrict__ S2adj, float* __restrict__ out) {
  __shared__ float As[64 * LDST];
  __shared__ float Bs[16 * LDST];

  const int tid  = threadIdx.x;
  const int lane = tid & 31;
  const int w    = tid >> 5;        // 0..3
  const int l    = lane & 15;
  const int half = lane >> 4;
  const int tm   = w * 16;
  const int m0   = blockIdx.x * 64;
  const int n0   = blockIdx.y * 16;

  const int ar = tid >> 3;          // 0..15
  const int ac = (tid & 7) << 2;    // 0..28
  const int brow = n0 + ar;         // class index for this B row
  const bool bvalid = (brow < 65);

  v8f acc0 = {}; v8f acc1 = {};

  for (int k0 = 0; k0 < 2048; k0 += 32) {
    const float4 av0 = *(const float4*)(z1 + (size_t)(m0 + ar) * 2048 + k0 + ac);
    const float4 av1 = *(const float4*)(z1 + (size_t)(m0 + ar + 16) * 2048 + k0 + ac);
    const float4 av2 = *(const float4*)(z1 + (size_t)(m0 + ar + 32) * 2048 + k0 + ac);
    const float4 av3 = *(const float4*)(z1 + (size_t)(m0 + ar + 48) * 2048 + k0 + ac);
    const float4 bv  = bvalid
        ? *(const float4*)(Wfc + (size_t)brow * 3072 + k0 + ac)
        : make_float4(0.0f, 0.0f, 0.0f, 0.0f);
    __syncthreads();
    st_lds4(&As[(ar)      * LDST + ac], av0);
    st_lds4(&As[(ar + 16) * LDST + ac], av1);
    st_lds4(&As[(ar + 32) * LDST + ac], av2);
    st_lds4(&As[(ar + 48) * LDST + ac], av3);
    st_lds4(&Bs[ar * LDST + ac], bv);
    __syncthreads();

    const float* ap = &As[(tm + l) * LDST + 2 * half];
    const float* bp = &Bs[l * LDST + 2 * half];
#pragma unroll
    for (int kk = 0; kk < 32; kk += 8) {
      v2f a0 = *(const v2f*)(ap + kk);
      v2f b0 = *(const v2f*)(bp + kk);
      acc0 = __builtin_amdgcn_wmma_f32_16x16x4_f32(false, a0, false, b0,
                                                   (short)0, acc0, false, false);
      v2f a1 = *(const v2f*)(ap + kk + 4);
      v2f b1 = *(const v2f*)(bp + kk + 4);
      acc1 = __builtin_amdgcn_wmma_f32_16x16x4_f32(false, a1, false, b1,
                                                   (short)0, acc1, false, false);
    }
  }

  const v8f acc = acc0 + acc1;
  const int c = n0 + l;
  if (c < 65) {
    const float s2 = S2adj[c];
#pragma unroll
    for (int r = 0; r < 8; ++r) {
      const int m = m0 + tm + r + 8 * half;
      out[(size_t)m * 65 + c] = 2048.0f * acc[r] + s2;
    }
  }
}

// ---------------------------------------------------------------------------
extern "C" void kernel_launch(void* const* d_in, const int* in_sizes, int n_in,
                              void* d_out, int out_size, void* d_ws, size_t ws_size,
                              hipStream_t stream) {
  const float* z1  = (const float*)d_in[0];   // [2048, 2048]
  const float* z2  = (const float*)d_in[1];   // [2048, 3*56*56] = [2048, 9408]
  const float* Wp  = (const float*)d_in[2];   // [1024, 9408]
  const float* bp  = (const float*)d_in[3];   // [1024]
  const float* Wfc = (const float*)d_in[4];   // [65, 3072]
  const float* bfc = (const float*)d_in[5];   // [65]
  float* out = (float*)d_out;                 // [2048, 65]

  float* zsum  = (float*)d_ws;                // 1024 floats
  float* S2adj = zsum + 1024;                 // 65 floats

  hipLaunchKernelGGL(k_zero, dim3(8), dim3(128), 0, stream, zsum);
  hipLaunchKernelGGL(k_proj_colsum, dim3(32, 8), dim3(512), 0, stream,
                     z2, Wp, bp, zsum);
  hipLaunchKernelGGL(k_s2, dim3(65), dim3(128), 0, stream,
                     zsum, Wfc, bfc, S2adj);
  hipLaunchKernelGGL(k_fc_out, dim3(32, 5), dim3(128), 0, stream,
                     z1, Wfc, S2adj, out);
}